// PoincarePlaneDistance_16750372454908
// MI455X (gfx1250) — compile-verified
//
#include <hip/hip_runtime.h>
#include <hip/hip_bf16.h>

// CDNA5 / gfx1250, wave32. Poincare plane-distance.
// WMMA f32 16x16x4 computes the full affine row-transform per tile:
//   A-matrix row m = [e0, e1, x2, 1] * r2   (r2 = 1/(1-x2), full K=4 used)
//   B col (tile 1) = [-2p0, -2p1, 1, 1]  ->  (1 - 2<x,p> + x2) * r2 = A*r2
//   B col (tile 2) = [ a0,   a1,  0, 0]  ->  <x,a> * r2
// Conformal identity (verified symbolically): 1 - ||mobius(-p,x)||^2
//   = (1-||p||^2)(1-||x||^2)/C, so the Mobius denominator C cancels and
//   z = 2*(Bm*xa - A*pa) / (Bm*(1-x2)*||a||) -- both reciprocals hoist.

typedef float v2f __attribute__((ext_vector_type(2)));
typedef float v8f __attribute__((ext_vector_type(8)));

namespace {
constexpr int   kS       = 64;     // spaces per sample
constexpr int   kP       = 256;    // planes
constexpr int   kB       = 4096;   // batch
constexpr float kMinNorm = 1e-15f;
constexpr float kMaxNorm = 1.0f - 4e-3f;   // geoopt ball projection radius
constexpr float kLn2     = 0.69314718055994531f;
}

__device__ __forceinline__ float fast_rcp(float x)  { return __builtin_amdgcn_rcpf(x); }
__device__ __forceinline__ float fast_sqrt(float x) { return __builtin_amdgcn_sqrtf(x); }
__device__ __forceinline__ float fast_log2(float x) { return __builtin_amdgcn_logf(x); }

__global__ __launch_bounds__(128) void poincare_plane_dist_wmma(
    const float* __restrict__ input,     // (B, S, 2)
    const float* __restrict__ normals,   // (P, S, 2)  = 'a'
    const float* __restrict__ bias,      // (P, S, 2)  = 'p'
    float* __restrict__ out)             // (B, P)
{
  const int tid  = threadIdx.x;
  const int lane = tid & 31;
  const int wave = tid >> 5;
  const int lo   = lane & 15;   // tile row (A) / tile col (B) id
  const int hi   = lane >> 4;   // lane half: supplies K=2,3 of A and B

  const int b_base = blockIdx.x << 4;                  // 16 samples per tile
  const int p_base = (blockIdx.y * 4 + wave) << 4;     // 16 planes per wave

  const int bl = b_base + lo;
  const int pl = p_base + lo;

  const float2* __restrict__ xin = reinterpret_cast<const float2*>(input)   + bl * kS;
  const float2* __restrict__ pin = reinterpret_cast<const float2*>(bias)    + pl * kS;
  const float2* __restrict__ ain = reinterpret_cast<const float2*>(normals) + pl * kS;

  v8f acc = {};

  for (int s = 0; s < kS; ++s) {
    // ---------- x side: expmap0 (tanh map onto ball + projection) ----------
    float2 u  = xin[s];
    float un2 = fmaf(u.x, u.x, u.y * u.y);
    float n   = fmaxf(fast_sqrt(un2), kMinNorm);
    float f   = tanhf(n) * fast_rcp(n);
    float y0  = u.x * f, y1 = u.y * f;
    float yn2 = fmaf(y0, y0, y1 * y1);
    float yn  = fmaxf(fast_sqrt(yn2), kMinNorm);
    float psc = (yn > kMaxNorm) ? (kMaxNorm * fast_rcp(yn)) : 1.0f;
    float e0  = y0 * psc, e1 = y1 * psc;
    float x2v = fmaf(e0, e0, e1 * e1);           // ||x||^2 (< 1 by projection)
    float r2  = fast_rcp(1.0f - x2v);            // 1/(1-x2)
    float e0r = e0 * r2, e1r = e1 * r2, x2r = x2v * r2;

    // ---------- plane side scalars for column p = pl ----------
    float2 pv = pin[s];
    float2 av = ain[s];
    float p2  = fmaf(pv.x, pv.x, pv.y * pv.y);
    float pa  = fmaf(pv.x, av.x, pv.y * av.y);
    float a2  = fmaf(av.x, av.x, av.y * av.y);
    float an  = fast_sqrt(fmaxf(a2, kMinNorm));  // ||a||
    float Bm  = 1.0f - p2;
    float r1  = fast_rcp(Bm * an);               // 1/(Bm*||a||)
    float c1  = (Bm + Bm) * r1;                  // coeff of xa*r2
    float c2  = (pa + pa) * r1;                  // coeff of A*r2
    float anl = an * kLn2;                       // ||a|| * ln2 (for log2->ln)
    float m2p0 = -(pv.x + pv.x);
    float m2p1 = -(pv.y + pv.y);

    // ---------- WMMA tiles ----------
    // A (16x4): lanes 0-15 rows M with K=0,1; lanes 16-31 rows M with K=2,3.
    // B (4x16): lanes 0-15 cols N with K=0,1; lanes 16-31 cols N with K=2,3.
    v2f Am, Bt, Ba;
    Am[0] = hi ? x2r  : e0r;  Am[1] = hi ? r2   : e1r;
    Bt[0] = hi ? 1.0f : m2p0; Bt[1] = hi ? 1.0f : m2p1;
    Ba[0] = hi ? 0.0f : av.x; Ba[1] = hi ? 0.0f : av.y;
    v8f zero = {};
    v8f A8  = __builtin_amdgcn_wmma_f32_16x16x4_f32(   // (1-2px+x2)*r2
        false, Am, false, Bt, (short)0, zero, false, false);
    v8f xa8 = __builtin_amdgcn_wmma_f32_16x16x4_f32(   // <x,a>*r2
        false, Am, false, Ba, (short)0, zero, false, false);

    // ---------- arcsinh tail: 6 VALU + 2 trans per output ----------
    #pragma unroll
    for (int v = 0; v < 8; ++v) {
      float z  = fmaf(c1, xa8[v], -(c2 * A8[v]));
      float w  = fast_sqrt(fmaf(z, z, 1.0f));
      float uu = fabsf(z) + w;                   // >= 1, safe for raw v_log
      float lg = fast_log2(uu);
      acc[v]   = fmaf(copysignf(anl, z), lg, acc[v]);
    }
  }

  #pragma unroll
  for (int v = 0; v < 8; ++v) {
    const int row = b_base + v + (hi << 3);      // D layout: M = v + 8*hi
    out[row * kP + p_base + lo] = acc[v];
  }
}

extern "C" void kernel_launch(void* const* d_in, const int* in_sizes, int n_in,
                              void* d_out, int out_size, void* d_ws, size_t ws_size,
                              hipStream_t stream) {
  (void)in_sizes; (void)n_in; (void)out_size; (void)d_ws; (void)ws_size;
  const float* input   = (const float*)d_in[0];   // (4096, 128)
  const float* normals = (const float*)d_in[1];   // (256, 64, 2)
  const float* bias    = (const float*)d_in[2];   // (256, 64, 2)
  float* out           = (float*)d_out;           // (4096, 256)

  dim3 block(128);                 // 4 waves, one 16-plane tile each
  dim3 grid(kB / 16, kP / 64);     // (256, 4)
  poincare_plane_dist_wmma<<<grid, block, 0, stream>>>(input, normals, bias, out);
}